// L1Attn_523986010475
// MI455X (gfx1250) — compile-verified
//
#include <hip/hip_runtime.h>
#include <stdint.h>

// Problem constants (B=1, S=1024, H=8, W=32), f32 in / f32 out.
#define S_DIM 1024
#define H_DIM 8
#define W_DIM 32
#define TS 64                 // s-tile per block
#define TT 64                 // t-tile per block
#define LDW 68                // TS + 4 floats pad -> 272B row stride (16B aligned, bank-friendly)

// CDNA5 async global->LDS path (ASYNCcnt-tracked), probed via __has_builtin so the
// source compiles on toolchains lacking the builtin.
#if defined(__HIP_DEVICE_COMPILE__) && defined(__gfx1250__) && \
    __has_builtin(__builtin_amdgcn_global_load_async_to_lds_b32) && \
    __has_builtin(__builtin_amdgcn_s_wait_asynccnt)
#define USE_ASYNC_LDS 1
#else
#define USE_ASYNC_LDS 0
#endif

// The builtin's parameters are address-space-qualified int pointers
// (clang diagnostic: "parameter of type '__device__ int *'").
typedef __attribute__((address_space(1))) int gbl_int_t;
typedef __attribute__((address_space(3))) int lds_int_t;

__device__ __forceinline__ void stage_f32_to_lds(const float* __restrict__ g,
                                                 float* __restrict__ l) {
#if USE_ASYNC_LDS
  // Per-lane LDS scatter address (low 32 bits of the generic LDS pointer are the
  // LDS byte offset; flat->LDS truncation per ISA 10.2).
  __builtin_amdgcn_global_load_async_to_lds_b32(
      (gbl_int_t*)(uintptr_t)g,
      (lds_int_t*)(uintptr_t)(uint32_t)(uintptr_t)l,
      /*offset=*/0, /*cpol=*/0);
#else
  *l = *g;
#endif
}

__global__ __launch_bounds__(256, 2) void l1attn_tile_kernel(
    const float* __restrict__ q, const float* __restrict__ k,
    float* __restrict__ out) {
  // Transposed tiles: [w][s_local] so the hot loop reads ds_load_b128.
  __shared__ float lq[W_DIM][LDW];
  __shared__ float lk[W_DIM][LDW];

  const int h   = blockIdx.x;
  const int s0  = blockIdx.y * TS;
  const int t0  = blockIdx.z * TT;
  const int tid = threadIdx.x;

  // ---- Stage q/k tiles (coalesced along w), transposed into LDS ----
  const int w  = tid & (W_DIM - 1);   // 0..31  -> contiguous 128B global reads per wave
  const int r0 = tid >> 5;            // 0..7

  const float* qb = q + (size_t)s0 * (H_DIM * W_DIM) + h * W_DIM + w;
  const float* kb = k + (size_t)t0 * (H_DIM * W_DIM) + h * W_DIM + w;

#pragma unroll
  for (int p = 0; p < TS / 8; ++p) {
    const int sl = r0 + p * 8;  // 0..63
    stage_f32_to_lds(qb + (size_t)sl * (H_DIM * W_DIM), &lq[w][sl]);
    stage_f32_to_lds(kb + (size_t)sl * (H_DIM * W_DIM), &lk[w][sl]);
  }
#if USE_ASYNC_LDS
  __builtin_amdgcn_s_wait_asynccnt(0);
#endif
  __syncthreads();

  // ---- Compute: 4x4 micro-tile per thread, loop over w ----
  const int tx = tid & 15;    // t micro-tile select
  const int ty = tid >> 4;    // s micro-tile select
  const int sb = ty * 4;
  const int tb = tx * 4;

  float acc[4][4];
#pragma unroll
  for (int i = 0; i < 4; ++i)
#pragma unroll
    for (int j = 0; j < 4; ++j) acc[i][j] = 0.0f;

#pragma unroll 8
  for (int ww = 0; ww < W_DIM; ++ww) {
    const float4 qv = *(const float4*)(&lq[ww][sb]);  // ds_load_b128 (broadcast in-wave)
    const float4 kv = *(const float4*)(&lk[ww][tb]);  // ds_load_b128 (64 banks, no conflict)
    const float qs[4] = {qv.x, qv.y, qv.z, qv.w};
    const float ks[4] = {kv.x, kv.y, kv.z, kv.w};
#pragma unroll
    for (int i = 0; i < 4; ++i)
#pragma unroll
      for (int j = 0; j < 4; ++j)
        acc[i][j] += __builtin_fabsf(qs[i] - ks[j]);  // v_sub_f32 + v_add_f32 |src|
  }

  // ---- Store: out[((s*1024)+t)*8 + h], write-once -> non-temporal ----
  const float nscale = -0.17677669529663687f;  // -1/sqrt(32)
#pragma unroll
  for (int i = 0; i < 4; ++i) {
    float* orow = out + ((size_t)(s0 + sb + i) * S_DIM + (t0 + tb)) * H_DIM + h;
#pragma unroll
    for (int j = 0; j < 4; ++j)
      __builtin_nontemporal_store(acc[i][j] * nscale, orow + (size_t)j * H_DIM);
  }
}

extern "C" void kernel_launch(void* const* d_in, const int* in_sizes, int n_in,
                              void* d_out, int out_size, void* d_ws, size_t ws_size,
                              hipStream_t stream) {
  (void)in_sizes; (void)n_in; (void)d_ws; (void)ws_size; (void)out_size;
  const float* q = (const float*)d_in[0];
  const float* k = (const float*)d_in[1];
  float* out = (float*)d_out;

  dim3 grid(H_DIM, S_DIM / TS, S_DIM / TT);  // (8, 16, 16)
  dim3 block(256);
  l1attn_tile_kernel<<<grid, block, 0, stream>>>(q, k, out);
}